// CalibratedCE2d_88493506167215
// MI455X (gfx1250) — compile-verified
//
#include <hip/hip_runtime.h>

// ---------------------------------------------------------------------------
// CalibratedCE2d on MI455X (gfx1250, wave32)
//   predict: (8, 21, 513, 513) f32 NCHW   target: (8,513,513) i32
//   confidence: (NHW,) f32                accuracies: (15,) f32   n_bin: i32
// Bandwidth-bound streaming logsumexp + weighted reduction.
// WMMA f32 16x16x4 used as a single-instruction 64-value cross-lane adder.
// ---------------------------------------------------------------------------

#define NN 8
#define CC 21
#define HH 513
#define WW 513
#define NBINS_TOT 15

constexpr int HW  = HH * WW;          // 263169
constexpr int NHW = NN * HW;          // 2105352

typedef __attribute__((ext_vector_type(2))) float v2f;
typedef __attribute__((ext_vector_type(8))) float v8f;

// Reduce a wave's 32 (a,b) float pairs to (sum_a, sum_b) broadcast to all
// lanes, using one V_WMMA_F32_16X16X4_F32:
//   A (16x4) rows 0..7  <- the 32 'a' values
//   A (16x4) rows 8..15 <- the 32 'b' values
//   B = ones, C = 0  =>  D[m][n] = rowsum(A,m) for every n.
// D layout: vgpr j, lanes 0..15 = row j (a-groups); lanes 16..31 = row j+8
// (b-groups). Summing the 8 D vgprs gives sum_a on lanes 0..15 and sum_b on
// lanes 16..31.
__device__ inline float2 wave_reduce_pair(float a, float b) {
    const int lane = threadIdx.x & 31;
    // A-matrix per-lane element sources:
    //   lane m      (0..7)  : (a[4m+0], a[4m+1])      row m
    //   lane m+8    (8..15) : (b[4m+0], b[4m+1])      row m+8
    //   lane m+16  (16..23) : (a[4m+2], a[4m+3])      row m   (K=2,3)
    //   lane m+24  (24..31) : (b[4m+2], b[4m+3])      row m+8 (K=2,3)
    const int g   = lane & 7;
    const int hi  = (lane & 16) ? 2 : 0;
    const int s0  = 4 * g + hi;
    const int s1  = s0 + 1;
    const float a0 = __shfl(a, s0, 32);
    const float a1 = __shfl(a, s1, 32);
    const float b0 = __shfl(b, s0, 32);
    const float b1 = __shfl(b, s1, 32);
    const bool useB = (lane & 8) != 0;
    v2f A;
    A[0] = useB ? b0 : a0;
    A[1] = useB ? b1 : a1;
    v2f B;
    B[0] = 1.0f;
    B[1] = 1.0f;
    v8f C = {};
    v8f D = __builtin_amdgcn_wmma_f32_16x16x4_f32(
        /*neg_a=*/false, A, /*neg_b=*/false, B,
        /*c_mod=*/(short)0, C, /*reuse_a=*/false, /*reuse_b=*/false);
    float r = D[0] + D[1] + D[2] + D[3] + D[4] + D[5] + D[6] + D[7];
    float sumA = __shfl(r, 0, 32);   // lanes 0..15 hold sum of rows 0..7
    float sumB = __shfl(r, 16, 32);  // lanes 16..31 hold sum of rows 8..15
    return make_float2(sumA, sumB);
}

__global__ __launch_bounds__(256) void cal_ce_main_kernel(
    const float* __restrict__ predict,
    const int*   __restrict__ target,
    const float* __restrict__ confidence,
    const float* __restrict__ accuracies,
    const int*   __restrict__ nbin_ptr,
    float*       __restrict__ partials,   // [2*nblocks]: lp sums then sel sums
    int nblocks)
{
    __shared__ float s_coeff[NBINS_TOT];
    __shared__ int   s_nbin;
    __shared__ float s_wlp[8], s_wsel[8];

    const int t = threadIdx.x;
    if (t < NBINS_TOT) {
        // coeff = acc*10 - (1-acc)*50 = acc*60 - 50
        s_coeff[t] = fmaf(accuracies[t], 60.0f, -50.0f);
    }
    if (t == 0) s_nbin = *nbin_ptr;
    __syncthreads();

    const int nb     = s_nbin;
    const int stride = nblocks * 256;

    float lp_acc  = 0.0f;
    float sel_acc = 0.0f;

    for (int i = blockIdx.x * 256 + t; i < NHW; i += stride) {
        const int n   = i / HW;
        const int rem = i - n * HW;
        const float* __restrict__ p = predict + (size_t)n * (CC * HW) + rem;

        // 21 independent coalesced loads -> registers (max MLP, one HBM pass)
        float x[CC];
#pragma unroll
        for (int c = 0; c < CC; ++c) x[c] = p[(size_t)c * HW];

        const int tg = target[i];

        float m = x[0];
#pragma unroll
        for (int c = 1; c < CC; ++c) m = fmaxf(m, x[c]);

        float xt = x[0];
#pragma unroll
        for (int c = 1; c < CC; ++c) xt = (c == tg) ? x[c] : xt;

        float s = 0.0f;
#pragma unroll
        for (int c = 0; c < CC; ++c) s += __expf(x[c] - m);

        const float lse   = m + __logf(s);
        const float logpt = xt - lse;

        const float cf = confidence[i];
        int bin = (int)ceilf(cf * (float)NBINS_TOT) - 1;
        bin = bin < 0 ? 0 : bin;
        const int bmax = nb - 1;
        bin = bin > bmax ? bmax : bin;

        const bool  valid = (cf > 0.0f) && (cf <= 1.0f);
        const float coeff = s_coeff[bin];
        const bool  sel   = valid && (coeff > 0.0f);

        lp_acc  += sel ? logpt * coeff : 0.0f;
        sel_acc += sel ? 1.0f : 0.0f;
    }

    // Wave-level reduce via WMMA (EXEC is all-ones here: no divergence).
    const float2 wr = wave_reduce_pair(lp_acc, sel_acc);
    const int wv   = t >> 5;
    const int lane = t & 31;
    if (lane == 0) { s_wlp[wv] = wr.x; s_wsel[wv] = wr.y; }
    __syncthreads();

    if (t == 0) {
        float l = 0.0f, sl = 0.0f;
#pragma unroll
        for (int w = 0; w < 8; ++w) { l += s_wlp[w]; sl += s_wsel[w]; }
        partials[blockIdx.x]           = l;
        partials[nblocks + blockIdx.x] = sl;
    }
}

// Single-wave deterministic finalize: fixed-order strided accumulation then
// one WMMA cross-lane reduction.  loss = -sum(lp) / sum(sel)
__global__ __launch_bounds__(32) void cal_ce_final_kernel(
    const float* __restrict__ partials, int nblocks, float* __restrict__ out)
{
    const int lane = threadIdx.x;
    float lp = 0.0f, sl = 0.0f;
    for (int j = lane; j < nblocks; j += 32) {
        lp += partials[j];
        sl += partials[nblocks + j];
    }
    const float2 r = wave_reduce_pair(lp, sl);
    if (lane == 0) out[0] = -r.x / r.y;
}

extern "C" void kernel_launch(void* const* d_in, const int* in_sizes, int n_in,
                              void* d_out, int out_size, void* d_ws, size_t ws_size,
                              hipStream_t stream) {
    (void)in_sizes; (void)n_in; (void)out_size;

    const float* predict    = (const float*)d_in[0];
    const int*   target     = (const int*)d_in[1];
    const float* confidence = (const float*)d_in[2];
    const float* accuracies = (const float*)d_in[3];
    const int*   nbin_ptr   = (const int*)d_in[4];
    float*       out        = (float*)d_out;
    float*       partials   = (float*)d_ws;

    int nblocks = 2048;
    const size_t need = (size_t)nblocks * 2 * sizeof(float);
    if (ws_size < need) {
        nblocks = (int)(ws_size / (2 * sizeof(float)));
        if (nblocks < 1) nblocks = 1;
    }

    cal_ce_main_kernel<<<nblocks, 256, 0, stream>>>(
        predict, target, confidence, accuracies, nbin_ptr, partials, nblocks);
    cal_ce_final_kernel<<<1, 32, 0, stream>>>(partials, nblocks, out);
}